// BarycentricCoordinates_23218593202561
// MI455X (gfx1250) — compile-verified
//
#include <hip/hip_runtime.h>
#include <stdint.h>

#define V_TOT 3000
#define NPTS  8
#define NTRI  56
#define RA    40
#define WPB   8      // waves (projections) per block
#define REC   19     // floats per triangle record (odd -> no LDS bank conflicts)

typedef float v2f __attribute__((ext_vector_type(2)));
typedef float v8f __attribute__((ext_vector_type(8)));
typedef unsigned int v4u __attribute__((ext_vector_type(4)));
typedef int v4i __attribute__((ext_vector_type(4)));
typedef int v8i __attribute__((ext_vector_type(8)));

__constant__ unsigned char TRI3[NTRI][3] = {
  {0,1,2},{0,1,3},{0,1,4},{0,1,5},{0,1,6},{0,1,7},
  {0,2,3},{0,2,4},{0,2,5},{0,2,6},{0,2,7},
  {0,3,4},{0,3,5},{0,3,6},{0,3,7},
  {0,4,5},{0,4,6},{0,4,7},
  {0,5,6},{0,5,7},
  {0,6,7},
  {1,2,3},{1,2,4},{1,2,5},{1,2,6},{1,2,7},
  {1,3,4},{1,3,5},{1,3,6},{1,3,7},
  {1,4,5},{1,4,6},{1,4,7},
  {1,5,6},{1,5,7},
  {1,6,7},
  {2,3,4},{2,3,5},{2,3,6},{2,3,7},
  {2,4,5},{2,4,6},{2,4,7},
  {2,5,6},{2,5,7},
  {2,6,7},
  {3,4,5},{3,4,6},{3,4,7},
  {3,5,6},{3,5,7},
  {3,6,7},
  {4,5,6},{4,5,7},
  {4,6,7},
  {5,6,7}
};

__global__ __launch_bounds__(256)
void BarycentricCoordinates_23218593202561_kernel(const float* __restrict__ tmpl,
                                                  const float* __restrict__ proj,
                                                  float* __restrict__ out)
{
    __shared__ float projLds[WPB * NPTS * 2];           // 128 floats, TDM destination
    __shared__ float tmplLds[RA * 2];                   // 80 floats
    __shared__ float recLds[WPB][NTRI * REC];           // per-wave triangle records

    const int lane = threadIdx.x & 31;
    const int wave = threadIdx.x >> 5;
    const int v    = blockIdx.x * WPB + wave;           // 375*8 == 3000 exactly

    // ---- TDM: async load of this block's 8 projections (128 f32) into LDS ----
    if (threadIdx.x < 32) {   // wave-uniform guard: wave 0 issues the DMA
        unsigned long long gaddr =
            (unsigned long long)(uintptr_t)(proj + (size_t)blockIdx.x * WPB * NPTS * 2);
        unsigned int ldsOff = (unsigned int)(uintptr_t)(&projLds[0]); // low 32b = LDS offset
        v4u g0;
        g0[0] = 1u;                                        // count=1 (valid descriptor)
        g0[1] = ldsOff;                                    // lds_addr
        g0[2] = (unsigned int)(gaddr & 0xFFFFFFFFu);       // global_addr[31:0]
        g0[3] = (unsigned int)((gaddr >> 32) & 0x1FFFFFFu) // global_addr[56:32]
              | 0x80000000u;                               // type=2 ("image")
        v8i g1;
        g1[0] = 0x00020000;          // workgroup_mask=0, data_size=2 (4B)
        g1[1] = (int)(128u << 16);   // tensor_dim0[15:0]=128 in bits[63:48]
        g1[2] = (int)(1u << 16);     // tensor_dim0 hi=0, tensor_dim1=1
        g1[3] = (int)(128u << 16);   // tensor_dim1 hi=0, tile_dim0=128
        g1[4] = 0;                   // tile_dim1=0, tile_dim2=0 (unused)
        g1[5] = 128;                 // tensor_dim0_stride lo
        g1[6] = 0;                   // stride0 hi, stride1 lo
        g1[7] = 0;                   // stride1 hi
        v4i z4 = {0, 0, 0, 0};
#if __has_include(<hip/amd_detail/amd_gfx1250_TDM.h>)
        v8i z8 = {0, 0, 0, 0, 0, 0, 0, 0};
        __builtin_amdgcn_tensor_load_to_lds(g0, g1, z4, z4, z8, 0);
#else
        __builtin_amdgcn_tensor_load_to_lds(g0, g1, z4, z4, 0);
#endif
        __builtin_amdgcn_s_wait_tensorcnt(0);
    }
    for (int i = threadIdx.x; i < RA * 2; i += 256) tmplLds[i] = tmpl[i];
    __syncthreads();

    // ---- Phase 1: per-triangle setup (f64 Delaunay predicate) ----
    const float* P  = &projLds[wave * NPTS * 2];
    float*       Rc = &recLds[wave][0];
    for (int t = lane; t < NTRI; t += 32) {
        const int i0 = TRI3[t][0], i1 = TRI3[t][1], i2 = TRI3[t][2];
        const float ax = P[2*i0], ay = P[2*i0+1];
        const float bx = P[2*i1], by = P[2*i1+1];
        const float cx = P[2*i2], cy = P[2*i2+1];
        // CCW angular sort == even/odd permutation == sign of orientation
        const double cr = ((double)bx - ax) * ((double)cy - ay)
                        - ((double)by - ay) * ((double)cx - ax);
        const double sgn = (cr > 0.0) ? 1.0 : -1.0;
        int flag = 0;
        for (int p = 0; p < NPTS; ++p) {
            const double px = P[2*p], py = P[2*p+1];
            const double x1 = (double)ax - px, y1 = (double)ay - py, z1 = x1*x1 + y1*y1;
            const double x2 = (double)bx - px, y2 = (double)by - py, z2 = x2*x2 + y2*y2;
            const double x3 = (double)cx - px, y3 = (double)cy - py, z3 = x3*x3 + y3*y3;
            const double det = x1*(y2*z3 - z2*y3) - y1*(x2*z3 - z2*x3) + z1*(x2*y3 - y2*x3);
            if (sgn * det > 0.0) flag = 1;   // some point inside circumcircle
        }
        const float v0x = cx - ax, v0y = cy - ay;   // C - A
        const float v1x = bx - ax, v1y = by - ay;   // B - A
        const float dot00 = v0x*v0x + v0y*v0y;
        const float dot01 = v0x*v1x + v0y*v1y;
        const float dot11 = v1x*v1x + v1y*v1y;
        const float denom = 1.0f / (dot00*dot11 - dot01*dot01);
        float* r = &Rc[t * REC];
        r[0] = v0x; r[1] = v0y; r[2] = -(v0x*ax + v0y*ay);
        r[3] = v1x; r[4] = v1y; r[5] = -(v1x*ax + v1y*ay);
        r[6] = dot00; r[7] = dot01; r[8] = dot11; r[9] = denom;
        r[10] = ax; r[11] = ay; r[12] = bx; r[13] = by; r[14] = cx; r[15] = cy;
        r[16] = flag ? 1.0f : 0.0f;
    }
    __syncthreads();

    // ---- Phase 2: WMMA f32 16x16x4 computes dot02/dot12 = [T|1] x [v; -v.A] ----
    const float INF = __builtin_inff();
    const int  col   = lane & 15;
    const bool hi    = lane >= 16;
    const int  rowHi = hi ? 8 : 0;

    // A fragments: rows = template points (Tx,Ty) with a ones column (K=2)
    v2f afr[3];
#pragma unroll
    for (int m = 0; m < 3; ++m) {
        const int ra = m * 16 + col;
        const bool ok = ra < RA;
        const float tx = ok ? tmplLds[2*ra]   : 0.0f;
        const float ty = ok ? tmplLds[2*ra+1] : 0.0f;
        v2f a;
        a[0] = hi ? (ok ? 1.0f : 0.0f) : tx;  // lanes>=16 hold K=2,3
        a[1] = hi ? 0.0f : ty;
        afr[m] = a;
    }
    // B fragments + per-lane triangle scalars, hoisted per n-tile
    v2f b02[4], b12[4];
    float rd00[4], rd01[4], rd11[4], rden[4];
    float rax[4], ray[4], rbx[4], rby[4], rcx[4], rcy[4];
    bool  rmask[4];
#pragma unroll
    for (int n = 0; n < 4; ++n) {
        const int t  = n * 16 + col;
        const bool ok = t < NTRI;
        const int tc = ok ? t : 0;
        const float* r = &Rc[tc * REC];
        float s0 = hi ? r[2] : r[0];
        float s1 = hi ? 0.0f : r[1];
        float u0 = hi ? r[5] : r[3];
        float u1 = hi ? 0.0f : r[4];
        s0 = ok ? s0 : 0.0f;  s1 = ok ? s1 : 0.0f;
        u0 = ok ? u0 : 0.0f;  u1 = ok ? u1 : 0.0f;
        v2f b; b[0] = s0; b[1] = s1; b02[n] = b;
        v2f c; c[0] = u0; c[1] = u1; b12[n] = c;
        rd00[n] = r[6]; rd01[n] = r[7]; rd11[n] = r[8]; rden[n] = r[9];
        rax[n] = r[10]; ray[n] = r[11]; rbx[n] = r[12];
        rby[n] = r[13]; rcx[n] = r[14]; rcy[n] = r[15];
        rmask[n] = (r[16] != 0.0f) || !ok;
    }

    const v8f cz = {0.f,0.f,0.f,0.f,0.f,0.f,0.f,0.f};

#pragma unroll
    for (int m = 0; m < 3; ++m) {
        float bd[8], bw0[8], bw1[8], bw2[8]; int bt[8];
        float rtx[8], rty[8];
#pragma unroll
        for (int j = 0; j < 8; ++j) {
            bd[j] = INF; bt[j] = 0; bw0[j] = bw1[j] = bw2[j] = 0.f;
            const int ra = m * 16 + rowHi + j;
            const int rc = (ra < RA) ? ra : 0;
            rtx[j] = tmplLds[2*rc]; rty[j] = tmplLds[2*rc+1];
        }

#pragma unroll
        for (int n = 0; n < 4; ++n) {
            const v8f d02 = __builtin_amdgcn_wmma_f32_16x16x4_f32(
                                false, afr[m], false, b02[n], (short)0, cz, false, false);
            const v8f d12 = __builtin_amdgcn_wmma_f32_16x16x4_f32(
                                false, afr[m], false, b12[n], (short)0, cz, false, false);
            const int t = n * 16 + col;

#pragma unroll
            for (int j = 0; j < 8; ++j) {
                const float dot02 = d02[j], dot12 = d12[j];
                float w2 = (rd11[n]*dot02 - rd01[n]*dot12) * rden[n];
                float w1 = (rd00[n]*dot12 - rd01[n]*dot02) * rden[n];
                float w0 = 1.0f - w2 - w1;
                w0 = (w0 != w0) ? -1.0f : w0;
                w1 = (w1 != w1) ? -1.0f : w1;
                w2 = (w2 != w2) ? -1.0f : w2;
                const bool cond = (w0 >= 1.0f) || (w0 <= 0.0f) ||
                                  (w1 >= 1.0f) || (w1 <= 0.0f) ||
                                  (w2 >= 1.0f) || (w2 <= 0.0f);
                const float tx = rtx[j], ty = rty[j];
                const float dist = sqrtf((rax[n]-tx)*(rax[n]-tx) + (ray[n]-ty)*(ray[n]-ty))
                                 + sqrtf((rbx[n]-tx)*(rbx[n]-tx) + (rby[n]-ty)*(rby[n]-ty))
                                 + sqrtf((rcx[n]-tx)*(rcx[n]-tx) + (rcy[n]-ty)*(rcy[n]-ty));
                const float d = (rmask[n] || cond) ? INF : dist;
                // branchless first-min update (t strictly increases with n)
                const bool take = d < bd[j];
                bd[j]  = take ? d  : bd[j];
                bt[j]  = take ? t  : bt[j];
                bw0[j] = take ? w0 : bw0[j];
                bw1[j] = take ? w1 : bw1[j];
                bw2[j] = take ? w2 : bw2[j];
            }
        }

        // argmin over triangle columns: 16-lane halves hold disjoint row groups
#pragma unroll
        for (int j = 0; j < 8; ++j) {
            float d = bd[j], w0 = bw0[j], w1 = bw1[j], w2 = bw2[j];
            int   ti = bt[j];
#pragma unroll
            for (int off = 8; off > 0; off >>= 1) {
                const float od = __shfl_xor(d,  off, 32);
                const int   ot = __shfl_xor(ti, off, 32);
                const float o0 = __shfl_xor(w0, off, 32);
                const float o1 = __shfl_xor(w1, off, 32);
                const float o2 = __shfl_xor(w2, off, 32);
                const bool take = (od < d) || (od == d && ot < ti);
                d  = take ? od : d;
                ti = take ? ot : ti;
                w0 = take ? o0 : w0;
                w1 = take ? o1 : w1;
                w2 = take ? o2 : w2;
            }
            if ((lane & 15) == 0) {
                const int ra = m * 16 + rowHi + j;
                if (ra < RA) {
                    const bool allm = !(d < INF);   // everything masked -> zeros
                    const float o0v = allm ? 0.0f : w0;
                    const float o1v = allm ? 0.0f : w1;
                    const float o2v = allm ? 0.0f : w2;
                    const int idx0 = allm ? 0 : (int)TRI3[ti][0];
                    const int idx1 = allm ? 0 : (int)TRI3[ti][1];
                    const int idx2 = allm ? 0 : (int)TRI3[ti][2];
                    const size_t base = ((size_t)v * RA + ra) * 3;
                    out[base + 0] = o0v; out[base + 1] = o1v; out[base + 2] = o2v;
                    const size_t ib = (size_t)V_TOT * RA * 3 + base;
                    out[ib + 0] = (float)idx0;
                    out[ib + 1] = (float)idx1;
                    out[ib + 2] = (float)idx2;
                }
            }
        }
    }
}

extern "C" void kernel_launch(void* const* d_in, const int* in_sizes, int n_in,
                              void* d_out, int out_size, void* d_ws, size_t ws_size,
                              hipStream_t stream) {
    (void)in_sizes; (void)n_in; (void)out_size; (void)d_ws; (void)ws_size;
    const float* tmpl = (const float*)d_in[0];   // (5,8,2) f32
    const float* proj = (const float*)d_in[1];   // (3000,8,2) f32
    float* out = (float*)d_out;                  // 360000 bc + 360000 idx
    dim3 grid(V_TOT / WPB), block(256);
    hipLaunchKernelGGL(BarycentricCoordinates_23218593202561_kernel,
                       grid, block, 0, stream, tmpl, proj, out);
}